// MaskRCNN_4191888081046
// MI455X (gfx1250) — compile-verified
//
#include <hip/hip_runtime.h>
#include <hip/hip_bf16.h>

// ROI-Align (Mask-RCNN multi-level crop_and_resize) for MI455X / gfx1250.
//
// One block (256 thr = 8 wave32) per box. Per-box sampling descriptors are
// precomputed once into LDS, stored per-(position, row-half) so each lane
// fetches exactly its own record with a single ds_load_b64 (no VALU selects).
// Each wave produces 16ch x 16pos output tiles with V_WMMA_F32_16X16X4_F32
// (K=4 = the four bilinear corners):
//   A (16x4): lanes 0-15  -> channel m, corners k=0,1 (row y0: x0, x1)
//             lanes 16-31 -> channel m, corners k=2,3 (row y1: x0, x1)
//   B (4x16): the 4 weights in column p, zero elsewhere
//   D (16x16) accumulates over the tile's 16 positions, stored per the
//   documented C/D layout (VGPR r -> rows M=r and M=r+8).
// 49 positions = 3 full 16-wide tiles (fully unrolled, no tail branches)
// + 1 leftover position handled by a tiny VALU epilogue.

typedef __attribute__((ext_vector_type(2))) float v2f;
typedef __attribute__((ext_vector_type(8))) float v8f;

#define POOL 7
#define NPOS 49          // 7*7
#define CCH  256

__global__ __launch_bounds__(256) void roialign_wmma_kernel(
    const float* __restrict__ boxes,
    const float* __restrict__ p2, const float* __restrict__ p3,
    const float* __restrict__ p4, const float* __restrict__ p5,
    float* __restrict__ out, int nboxes)
{
    const int n = blockIdx.x;
    if (n >= nboxes) return;

    // [position][half]: half 0 = row y0 taps (x0,x1), half 1 = row y1 taps.
    __shared__ int2   sO[NPOS * 2];   // corner offsets {y*H+x0, y*H+x1}
    __shared__ float2 sW[NPOS * 2];   // weights {w_x0, w_x1} (validity folded)

    const int tid = threadIdx.x;

    // ---- box params (uniform across block) ----
    const float by1 = boxes[n * 4 + 0];
    const float bx1 = boxes[n * 4 + 1];
    const float by2 = boxes[n * 4 + 2];
    const float bx2 = boxes[n * 4 + 3];
    const float bh = by2 - by1;
    const float bw = bx2 - bx1;

    // roi_level = clip(round(4 + log2(sqrt(h*w) / (224/1024))), 2, 5)
    const float t   = sqrtf(fmaxf(bh * bw, 1e-12f)) * (1024.0f / 224.0f);
    int lvl = (int)rintf(4.0f + log2f(t));   // rintf == round-half-even (jnp.round)
    lvl = lvl < 2 ? 2 : (lvl > 5 ? 5 : lvl);

    const int H = 256 >> (lvl - 2);          // square levels: W == H
    const float* __restrict__ fmap =
        (lvl == 2) ? p2 : (lvl == 3) ? p3 : (lvl == 4) ? p4 : p5;

    // ---- per-position descriptors into LDS (reused by all 256 channels) ----
    if (tid < NPOS) {
        const int j = tid / POOL;
        const int i = tid - j * POOL;
        const float hm1   = (float)(H - 1);
        const float scale = hm1 / 6.0f;

        const float ys  = by1 * hm1 + bh * scale * (float)j;
        const float y0f = floorf(ys);
        const float fy  = ys - y0f;
        int y0 = (int)y0f;
        y0 = y0 < 0 ? 0 : (y0 > H - 1 ? H - 1 : y0);
        const int y1 = (y0 + 1 > H - 1) ? (H - 1) : (y0 + 1);
        const float vy = (ys >= 0.0f && ys <= hm1) ? 1.0f : 0.0f;

        const float xs  = bx1 * hm1 + bw * scale * (float)i;
        const float x0f = floorf(xs);
        const float fx  = xs - x0f;
        int x0 = (int)x0f;
        x0 = x0 < 0 ? 0 : (x0 > H - 1 ? H - 1 : x0);
        const int x1 = (x0 + 1 > H - 1) ? (H - 1) : (x0 + 1);
        const float vx = (xs >= 0.0f && xs <= hm1) ? 1.0f : 0.0f;

        const float vm = vy * vx;            // masks invalid samples to exactly 0
        // half 0: row y0 -> corners k=0 (x0), k=1 (x1)
        sW[2 * tid + 0] = make_float2((1.f - fy) * (1.f - fx) * vm,
                                      (1.f - fy) * fx         * vm);
        sO[2 * tid + 0] = make_int2(y0 * H + x0, y0 * H + x1);
        // half 1: row y1 -> corners k=2 (x0), k=3 (x1)
        sW[2 * tid + 1] = make_float2(fy * (1.f - fx) * vm,
                                      fy * fx         * vm);
        sO[2 * tid + 1] = make_int2(y1 * H + x0, y1 * H + x1);
    }
    __syncthreads();

    const int wave = tid >> 5;               // 0..7
    const int lane = tid & 31;
    const int m    = lane & 15;              // channel-in-tile (A) / position (D store)
    const int half = lane >> 4;              // 0: corners k=0,1 (row y0); 1: k=2,3 (row y1)

    const size_t HW = (size_t)H * (size_t)H;

    // 48 full tiles per box: 16 channel-tiles x 3 position-tiles; 6 per wave.
    for (int s = 0; s < 6; ++s) {
        const int tile  = wave * 6 + s;
        const int ctile = tile / 3;
        const int ptile = tile - ctile * 3;
        const int c0 = ctile * 16;
        const int p0 = ptile * 16;

        const float* __restrict__ chbase = fmap + (size_t)(c0 + m) * HW;
        // warm the gather region (global_prefetch path)
        __builtin_prefetch(chbase + sO[2 * p0].x, 0, 0);

        v8f acc = {0.f, 0.f, 0.f, 0.f, 0.f, 0.f, 0.f, 0.f};

        #pragma unroll
        for (int k = 0; k < 16; ++k) {       // full trip: p0+k <= 47 always valid
            const int rec = 2 * (p0 + k) + half;   // lane-addressed descriptor
            const int2   o = sO[rec];              // ds_load_b64 broadcast per half
            const float2 w = sW[rec];

            // A: this lane's two corner taps (contiguous x0,x1 in its row)
            v2f a;
            a.x = chbase[o.x];
            a.y = chbase[o.y];

            // B: weights live only in column k (k is a literal after unroll)
            v2f b;
            b.x = (m == k) ? w.x : 0.0f;
            b.y = (m == k) ? w.y : 0.0f;

            acc = __builtin_amdgcn_wmma_f32_16x16x4_f32(
                false, a, false, b, (short)0, acc, false, false);
        }

        // D store: VGPR r holds channel (c0 + r + half*8), position (p0 + m).
        // pos <= 47 always -> unconditional, clause-friendly strided stores.
        const size_t obase =
            ((size_t)n * CCH + (size_t)(c0 + half * 8)) * NPOS + (size_t)(p0 + m);
        #pragma unroll
        for (int r = 0; r < 8; ++r) {
            out[obase + (size_t)r * NPOS] = acc[r];
        }
    }

    // ---- leftover position 48 (j=6, i=6): one channel per thread ----
    {
        const int c = tid;                   // 256 threads == 256 channels
        const float* __restrict__ cb = fmap + (size_t)c * HW;
        const int2   o0 = sO[2 * (NPOS - 1) + 0];
        const int2   o1 = sO[2 * (NPOS - 1) + 1];
        const float2 w0 = sW[2 * (NPOS - 1) + 0];
        const float2 w1 = sW[2 * (NPOS - 1) + 1];
        const float r = cb[o0.x] * w0.x + cb[o0.y] * w0.y
                      + cb[o1.x] * w1.x + cb[o1.y] * w1.y;
        out[((size_t)n * CCH + (size_t)c) * NPOS + (NPOS - 1)] = r;
    }
}

extern "C" void kernel_launch(void* const* d_in, const int* in_sizes, int n_in,
                              void* d_out, int out_size, void* d_ws, size_t ws_size,
                              hipStream_t stream) {
    const float* boxes = (const float*)d_in[0];
    const float* p2    = (const float*)d_in[1];
    const float* p3    = (const float*)d_in[2];
    const float* p4    = (const float*)d_in[3];
    const float* p5    = (const float*)d_in[4];
    float* out = (float*)d_out;

    const int nboxes = in_sizes[0] / 4;
    roialign_wmma_kernel<<<nboxes, 256, 0, stream>>>(boxes, p2, p3, p4, p5, out, nboxes);
}